// Centered1x1LC_87076166959474
// MI455X (gfx1250) — compile-verified
//
#include <hip/hip_runtime.h>

typedef float v2f __attribute__((ext_vector_type(2)));
typedef float v8f __attribute__((ext_vector_type(8)));
typedef int   v4i __attribute__((vector_size(16)));

typedef __attribute__((address_space(1))) v4i* gv4i_p;   // global int4*
typedef __attribute__((address_space(3))) v4i* sv4i_p;   // LDS int4*

#define HW   2304      // H*W
#define PX   4         // pixels per workgroup
#define KC   16        // K-chunk staged in LDS
#define NT   256       // threads per workgroup (8 waves)

#define GPTR(p) ((gv4i_p)(p))
#define SPTR(p) ((sv4i_p)(p))

#if __has_builtin(__builtin_amdgcn_global_load_async_to_lds_b128)
#define HAVE_ASYNC_LDS 1
#else
#define HAVE_ASYNC_LDS 0
#endif

__device__ __forceinline__ void wait_async_lds() {
#if HAVE_ASYNC_LDS
#if __has_builtin(__builtin_amdgcn_s_wait_asynccnt)
    __builtin_amdgcn_s_wait_asynccnt(0);
#else
    asm volatile("s_wait_asynccnt 0" ::: "memory");
#endif
#endif
}

// ---------------------------------------------------------------------------
// Main kernel: per-pixel GEMMs C_sum = (x+b) x W  and  C_b = b x W.
// Workgroup = 4 consecutive pixels; wave (of 8) = (pixel, j-half).
// ---------------------------------------------------------------------------
__global__ __launch_bounds__(NT) void lc_main(const float* __restrict__ xg,
                                              const float* __restrict__ bg,
                                              const float* __restrict__ wg,
                                              float* __restrict__ outg,
                                              float* __restrict__ wsg)
{
    __shared__ float lds[16384];            // 64 KB, phase-multiplexed
    float* W_s  = lds;                      // [KC][128][PX]  = 8192 f
    float* Ax_s = lds + 8192;               // [32][KC][PX]   = 2048 f  (x)
    float* Ab_s = lds + 10240;              // [32][KC][PX]   = 2048 f  (b)

    const int t    = threadIdx.x;
    const int lane = t & 31;
    const int wid  = t >> 5;
    const int px   = wid >> 1;              // pixel slot 0..3
    const int nh   = wid & 1;               // j half 0..1
    const int p0   = blockIdx.x * PX;       // first pixel of this block

    const int laneM = lane & 15;            // M (A) / N (B,C,D) index
    const int laneK = (lane >> 4) << 1;     // K offset 0 or 2 (A & B frags)
    const int laneR = (lane >> 4) << 3;     // row offset 0 or 8 (C/D tiles)

    v8f csum[2][4], cb[2][4];
    const v8f vzero = {0.f,0.f,0.f,0.f,0.f,0.f,0.f,0.f};
    #pragma unroll
    for (int m = 0; m < 2; ++m)
        #pragma unroll
        for (int n = 0; n < 4; ++n) { csum[m][n] = vzero; cb[m][n] = vzero; }

    for (int k0 = 0; k0 < 128; k0 += KC) {
        // ---- stage W chunk (16x128x4px) and raw x / b chunks (32x16x4px) ----
#if HAVE_ASYNC_LDS
        #pragma unroll
        for (int r = 0; r < 8; ++r) {
            int q = t + r * NT;                       // 0..2047 = (i,j)
            int i = q >> 7, j = q & 127;
            __builtin_amdgcn_global_load_async_to_lds_b128(
                GPTR(wg + (size_t)((k0 + i) * 128 + j) * HW + p0),
                SPTR(W_s + q * 4), 0, 0);
        }
        #pragma unroll
        for (int r = 0; r < 2; ++r) {
            int q  = t + r * NT;                      // 0..511 = (batch,i)
            int bb = q >> 4, i = q & 15;
            size_t g = (size_t)(bb * 128 + (k0 + i)) * HW + p0;
            __builtin_amdgcn_global_load_async_to_lds_b128(
                GPTR(xg + g), SPTR(Ax_s + q * 4), 0, 0);
            __builtin_amdgcn_global_load_async_to_lds_b128(
                GPTR(bg + g), SPTR(Ab_s + q * 4), 0, 0);
        }
        wait_async_lds();
#else
        #pragma unroll
        for (int r = 0; r < 8; ++r) {
            int q = t + r * NT;
            int i = q >> 7, j = q & 127;
            float4 v = *(const float4*)(wg + (size_t)((k0 + i) * 128 + j) * HW + p0);
            *(float4*)(W_s + q * 4) = v;
        }
        #pragma unroll
        for (int r = 0; r < 2; ++r) {
            int q  = t + r * NT;
            int bb = q >> 4, i = q & 15;
            size_t g = (size_t)(bb * 128 + (k0 + i)) * HW + p0;
            *(float4*)(Ax_s + q * 4) = *(const float4*)(xg + g);
            *(float4*)(Ab_s + q * 4) = *(const float4*)(bg + g);
        }
#endif
        __syncthreads();

        // ---- compute: 4 k-steps x 4 n-tiles x 2 m-tiles x 2 paths ----
        #pragma unroll
        for (int s = 0; s < 4; ++s) {
            int kl = 4 * s + laneK;
            v2f af[2], ab[2];
            #pragma unroll
            for (int m = 0; m < 2; ++m) {
                int o = ((16 * m + laneM) * KC + kl) * PX + px;
                float x0 = Ax_s[o], x1 = Ax_s[o + PX];
                float b0 = Ab_s[o], b1 = Ab_s[o + PX];
                af[m][0] = x0 + b0; af[m][1] = x1 + b1;   // (x+b) fragment
                ab[m][0] = b0;      ab[m][1] = b1;        // b fragment
            }
            #pragma unroll
            for (int nn = 0; nn < 4; ++nn) {
                int j = (nh * 4 + nn) * 16 + laneM;
                int o = (kl * 128 + j) * PX + px;
                v2f bf;
                bf[0] = W_s[o];
                bf[1] = W_s[o + 128 * PX];
                #pragma unroll
                for (int m = 0; m < 2; ++m) {
                    csum[m][nn] = __builtin_amdgcn_wmma_f32_16x16x4_f32(
                        false, af[m], false, bf, (short)0, csum[m][nn], false, false);
                    cb[m][nn] = __builtin_amdgcn_wmma_f32_16x16x4_f32(
                        false, ab[m], false, bf, (short)0, cb[m][nn], false, false);
                }
            }
        }
        __syncthreads();
    }

    // ---- stage C_sum -> LDS [b][j][px], then coalesced float4 stores ----
    #pragma unroll
    for (int m = 0; m < 2; ++m)
        #pragma unroll
        for (int nn = 0; nn < 4; ++nn) {
            int j = (nh * 4 + nn) * 16 + laneM;
            #pragma unroll
            for (int v = 0; v < 8; ++v) {
                int bb = 16 * m + laneR + v;
                lds[(bb * 128 + j) * 4 + px] = csum[m][nn][v];
            }
        }
    __syncthreads();
    #pragma unroll
    for (int r = 0; r < 16; ++r) {
        int q = t + r * NT;                           // (b*128 + j)
        *(float4*)(outg + (size_t)q * HW + p0) = *(const float4*)(lds + q * 4);
    }
    __syncthreads();

    // ---- stage C_b -> LDS [px][b][j], reduce over px, atomic to ws ----
    #pragma unroll
    for (int m = 0; m < 2; ++m)
        #pragma unroll
        for (int nn = 0; nn < 4; ++nn) {
            int j = (nh * 4 + nn) * 16 + laneM;
            #pragma unroll
            for (int v = 0; v < 8; ++v) {
                int bb = 16 * m + laneR + v;
                lds[px * 4096 + bb * 128 + j] = cb[m][nn][v];
            }
        }
    __syncthreads();
    #pragma unroll
    for (int r = 0; r < 16; ++r) {
        int q = t + r * NT;
        float v = lds[q] + lds[4096 + q] + lds[8192 + q] + lds[12288 + q];
        atomicAdd(wsg + q, v);
    }
}

// ---------------------------------------------------------------------------
__global__ void zero_ws(float* ws) {
    ws[blockIdx.x * 256 + threadIdx.x] = 0.0f;       // 16 blocks -> 4096 f
}

__global__ void mean_sub(float* __restrict__ outg, const float* __restrict__ ws) {
    const int bj = blockIdx.x;                       // 0..4095 = (b*128+j)
    const float m = ws[bj] * (1.0f / (float)HW);
    float* o = outg + (size_t)bj * HW;
    for (int p = threadIdx.x; p < HW; p += 256)
        o[p] -= m;
}

// ---------------------------------------------------------------------------
extern "C" void kernel_launch(void* const* d_in, const int* in_sizes, int n_in,
                              void* d_out, int out_size, void* d_ws, size_t ws_size,
                              hipStream_t stream) {
    const float* x = (const float*)d_in[0];   // (32,1,128,48,48,1,1)
    const float* b = (const float*)d_in[1];   // (32,1,128,48,48,1,1)
    const float* w = (const float*)d_in[2];   // (128,1,128,48,48,1,1)
    float* out = (float*)d_out;               // (32,1,128,48,48)
    float* ws  = (float*)d_ws;                // 4096 f32 accumulator

    zero_ws<<<16, 256, 0, stream>>>(ws);
    lc_main<<<HW / PX, NT, 0, stream>>>(x, b, w, out, ws);
    mean_sub<<<4096, 256, 0, stream>>>(out, ws);
}